// DifferentiableSSN_6700148982142
// MI455X (gfx1250) — compile-verified
//
#include <hip/hip_runtime.h>

typedef __attribute__((ext_vector_type(2))) float v2f;
typedef __attribute__((ext_vector_type(8))) float v8f;

#define DEV __device__ __forceinline__

namespace {
constexpr int kB  = 4;     // batch
constexpr int kC  = 16;    // feature channels
constexpr int kH  = 256;
constexpr int kW  = 256;
constexpr int kHW = kH * kW;        // 65536 pixels per image
constexpr int kM  = 256;            // superpixels
constexpr int kD  = 18;             // C + 2 (x, y)
constexpr int kDP = 20;             // D padded to multiple of 4 (WMMA K-step)
constexpr int WAVES_PER_BLOCK = 8;  // 256 threads
constexpr int PIX_PER_WAVE    = 16;
constexpr int ASSIGN_BLOCKS = (kB * kHW) / (WAVES_PER_BLOCK * PIX_PER_WAVE); // 2048
}

// p[n][k]: k<16 -> feature channel, 16 -> x = (n%W)/W, 17 -> y = (n/W)/H, >=18 -> 0 pad
DEV float pix_val(const float* __restrict__ F, int b, int k, int n) {
  if (k < kC)     return F[(size_t)(b * kC + k) * kHW + n];
  if (k == kC)    return (float)(n & (kW - 1)) * (1.0f / kW);
  if (k == kC + 1) return (float)(n >> 8) * (1.0f / kH);
  return 0.0f;
}

DEV v8f wmma4(v2f a, v2f b, v8f c) {
  // D = A(16x4 f32) * B(4x16 f32) + C(16x16 f32)
  return __builtin_amdgcn_wmma_f32_16x16x4_f32(false, a, false, b, (short)0, c,
                                               false, false);
}

// dot-product tile: 16 pixels x 16 centers, K = 20 (padded) via 5 WMMA k-steps.
// A layout: lane<16 holds (row=lane, k=4i+{0,1}); lane>=16 holds (row=lane-16, k=4i+{2,3})
// B layout: lane<16 holds (k=4i+{0,1}, col=lane); lane>=16 holds (k=4i+{2,3}, col=lane-16)
DEV v8f dot_tile(const v2f A[5], const float* __restrict__ ctr, int bm_base,
                 int lp, int hi) {
  const float* crow = ctr + (size_t)(bm_base + lp) * kDP;
  v8f acc = {};
#pragma unroll
  for (int i = 0; i < 5; ++i) {
    v2f bv = *(const v2f*)(crow + 4 * i + 2 * hi);  // 8B aligned (kDP even, k0 even)
    acc = wmma4(A[i], bv, acc);
  }
  return acc;
}

// e = exp(-sqrt(max(||p||^2 + ||c||^2 - 2 p.c, 1e-12)))
// D-layout: element (lane, v) is (row = v + 8*hi, col = lane&15)
DEV v8f exp_tile(v8f dot, float pn, float cnv, int hi) {
  v8f e;
#pragma unroll
  for (int v = 0; v < 8; ++v) {
    float pr = __shfl(pn, v + 8 * hi, 32);  // pn of pixel row
    float d2 = fmaxf(pr + cnv - 2.0f * dot[v], 1e-12f);
    e[v] = __expf(-sqrtf(d2));
  }
  return e;
}

// ---------------- initial centers: 16x16 block-mean pooling -----------------
__global__ void k_init_centers(const float* __restrict__ F,
                               float* __restrict__ ctr,
                               float* __restrict__ cn) {
  const int bm = blockIdx.x;        // b*256 + m
  const int b = bm >> 8, m = bm & 255;
  const int sy = m >> 4, sx = m & 15;
  const int d = threadIdx.x;        // one wave: lane = dim
  float s = 0.0f;
  if (d < kD) {
    for (int p = 0; p < 256; ++p) {
      const int h = sy * 16 + (p >> 4);
      const int w = sx * 16 + (p & 15);
      s += pix_val(F, b, d, h * kW + w);
    }
    s *= (1.0f / 256.0f);
  }
  if (d < kDP) ctr[(size_t)bm * kDP + d] = s;  // d=18,19 pad with 0
  float sq = s * s;                             // 0 for d >= 18
#pragma unroll
  for (int msk = 16; msk >= 1; msk >>= 1) sq += __shfl_xor(sq, msk, 32);
  if (d == 0) cn[bm] = sq;                      // ||c||^2
}

__global__ void k_zero(float* __restrict__ p, int n) {
  int i = blockIdx.x * blockDim.x + threadIdx.x;
  if (i < n) p[i] = 0.0f;
}

// c = (sum Q p) / (sum Q + 1e-6); also refresh ||c||^2 and zero pads
__global__ void k_update_centers(const float* __restrict__ acc,
                                 float* __restrict__ ctr,
                                 float* __restrict__ cn) {
  int i = blockIdx.x * blockDim.x + threadIdx.x;
  if (i >= kB * kM) return;
  const float* a = acc + (size_t)i * kDP;
  float* c = ctr + (size_t)i * kDP;
  const float inv = 1.0f / (a[18] + 1e-6f);  // a[18] = sum Q
  float sq = 0.0f;
#pragma unroll
  for (int d = 0; d < kD; ++d) {
    float v = a[d] * inv;
    c[d] = v;
    sq += v * v;
  }
  c[18] = 0.0f;
  c[19] = 0.0f;
  cn[i] = sq;
}

// centers_feat[b][c][m] = ctr[b][m][c], c < 16
__global__ void k_centers_feat(const float* __restrict__ ctr,
                               float* __restrict__ out2) {
  int i = blockIdx.x * blockDim.x + threadIdx.x;
  if (i >= kB * kC * kM) return;
  const int m = i & 255, bc = i >> 8, c = bc & 15, b = bc >> 4;
  out2[i] = ctr[(size_t)(b * kM + m) * kDP + c];
}

// -------- fused assignment + center accumulation (iterations 0 and 1) -------
__global__ void __launch_bounds__(256) k_assign_accum(
    const float* __restrict__ F, const float* __restrict__ ctr,
    const float* __restrict__ cn, float* __restrict__ accG) {
  __shared__ float lacc[kM * kDP];  // 20 KB block-local accumulator
  for (int i = threadIdx.x; i < kM * kDP; i += 256) lacc[i] = 0.0f;
  __syncthreads();

  const int lane = threadIdx.x & 31;
  const int lp = lane & 15, hi = lane >> 4;
  const int gw = blockIdx.x * WAVES_PER_BLOCK + (threadIdx.x >> 5);
  const int pix0 = gw * PIX_PER_WAVE;
  const int b = pix0 >> 16;             // / 65536
  const int nbase = pix0 & (kHW - 1);

  // A operand: 16 pixels x K (coalesced channel-major loads)
  v2f A[5];
#pragma unroll
  for (int i = 0; i < 5; ++i) {
    const int k0 = 4 * i + 2 * hi;
    A[i].x = pix_val(F, b, k0, nbase + lp);
    A[i].y = pix_val(F, b, k0 + 1, nbase + lp);
  }
  float pp = 0.0f;
#pragma unroll
  for (int i = 0; i < 5; ++i) pp += A[i].x * A[i].x + A[i].y * A[i].y;
  const float pn = pp + __shfl_xor(pp, 16, 32);  // halves hold complementary k's

  // B operands for Q^T * P: P in B-layout (k = pixel, col = dim)
  // B2a: dims 0..15 (features). B2b: col0=x, col1=y, col2=1 (-> sum Q), rest 0.
  v2f B2a[4], B2b[4];
#pragma unroll
  for (int j = 0; j < 4; ++j) {
    const int p0 = 4 * j + 2 * hi;
    const int n0 = nbase + p0, n1 = n0 + 1;
    B2a[j].x = F[(size_t)(b * kC + lp) * kHW + n0];
    B2a[j].y = F[(size_t)(b * kC + lp) * kHW + n1];
    float e0 = 0.0f, e1 = 0.0f;
    if (lp == 0) { e0 = (float)(n0 & 255) * (1.0f / 256); e1 = (float)(n1 & 255) * (1.0f / 256); }
    else if (lp == 1) { e0 = (float)(n0 >> 8) * (1.0f / 256); e1 = (float)(n1 >> 8) * (1.0f / 256); }
    else if (lp == 2) { e0 = 1.0f; e1 = 1.0f; }
    B2b[j].x = e0; B2b[j].y = e1;
  }

  // pass 1: exp(-dist) tiles + row sums over all 256 centers
  v8f E[16];
  float rp[8];
#pragma unroll
  for (int v = 0; v < 8; ++v) rp[v] = 0.0f;
#pragma unroll
  for (int ct = 0; ct < 16; ++ct) {
    const int bm_base = b * kM + ct * 16;
    const float cnv = cn[bm_base + lp];
    v8f e = exp_tile(dot_tile(A, ctr, bm_base, lp, hi), pn, cnv, hi);
#pragma unroll
    for (int v = 0; v < 8; ++v) rp[v] += e[v];
    E[ct] = e;
  }
  float inv[8];
#pragma unroll
  for (int v = 0; v < 8; ++v) {
    float s = rp[v];
    s += __shfl_xor(s, 1, 32);
    s += __shfl_xor(s, 2, 32);
    s += __shfl_xor(s, 4, 32);
    s += __shfl_xor(s, 8, 32);   // masks < 16: stays within the 16-lane half
    inv[v] = 1.0f / s;
  }

  // pass 2: normalize Q, accumulate Q^T P (WMMA) into LDS
#pragma unroll
  for (int ct = 0; ct < 16; ++ct) {
    v8f q;
#pragma unroll
    for (int v = 0; v < 8; ++v) q[v] = E[ct][v] * inv[v];
    v8f a0 = {}, a1 = {};
#pragma unroll
    for (int j = 0; j < 4; ++j) {
      // shuffle Q (D-layout) into Q^T (A-layout) for pixel chunk 4j..4j+3
      const int hj = (4 * j) >> 3;
      const int src = lp + (hj << 4);
      const float lo0 = __shfl(q[(4 * j) & 7], src, 32);
      const float g0  = __shfl(q[(4 * j + 2) & 7], src, 32);
      const float lo1 = __shfl(q[(4 * j + 1) & 7], src, 32);
      const float g1  = __shfl(q[(4 * j + 3) & 7], src, 32);
      v2f a2;
      a2.x = hi ? g0 : lo0;
      a2.y = hi ? g1 : lo1;
      a0 = wmma4(a2, B2a[j], a0);  // centers x dims 0..15
      a1 = wmma4(a2, B2b[j], a1);  // centers x {x, y, sumQ}
    }
#pragma unroll
    for (int v = 0; v < 8; ++v) {
      const int cidx = ct * 16 + v + 8 * hi;   // center row in D-layout
      atomicAdd(&lacc[cidx * kDP + lp], a0[v]);           // dims 0..15
      if (lp < 3) atomicAdd(&lacc[cidx * kDP + 16 + lp], a1[v]);  // 16,17,18=sumQ
    }
  }

  __syncthreads();
  const int bofs = b * kM * kDP;  // whole block is in one batch (128 px aligned)
  for (int i = threadIdx.x; i < kM * kDP; i += 256)
    atomicAdd(accG + bofs + i, lacc[i]);
}

// ------------- final assignment: stream softmax Q out (iteration 2) ---------
__global__ void __launch_bounds__(256) k_assign_write(
    const float* __restrict__ F, const float* __restrict__ ctr,
    const float* __restrict__ cn, float* __restrict__ Q) {
  const int lane = threadIdx.x & 31;
  const int lp = lane & 15, hi = lane >> 4;
  const int gw = blockIdx.x * WAVES_PER_BLOCK + (threadIdx.x >> 5);
  const int pix0 = gw * PIX_PER_WAVE;
  const int b = pix0 >> 16;
  const int nbase = pix0 & (kHW - 1);

  v2f A[5];
#pragma unroll
  for (int i = 0; i < 5; ++i) {
    const int k0 = 4 * i + 2 * hi;
    A[i].x = pix_val(F, b, k0, nbase + lp);
    A[i].y = pix_val(F, b, k0 + 1, nbase + lp);
  }
  float pp = 0.0f;
#pragma unroll
  for (int i = 0; i < 5; ++i) pp += A[i].x * A[i].x + A[i].y * A[i].y;
  const float pn = pp + __shfl_xor(pp, 16, 32);

  v8f E[16];
  float rp[8];
#pragma unroll
  for (int v = 0; v < 8; ++v) rp[v] = 0.0f;
#pragma unroll
  for (int ct = 0; ct < 16; ++ct) {
    const int bm_base = b * kM + ct * 16;
    const float cnv = cn[bm_base + lp];
    v8f e = exp_tile(dot_tile(A, ctr, bm_base, lp, hi), pn, cnv, hi);
#pragma unroll
    for (int v = 0; v < 8; ++v) rp[v] += e[v];
    E[ct] = e;
  }
  float inv[8];
#pragma unroll
  for (int v = 0; v < 8; ++v) {
    float s = rp[v];
    s += __shfl_xor(s, 1, 32);
    s += __shfl_xor(s, 2, 32);
    s += __shfl_xor(s, 4, 32);
    s += __shfl_xor(s, 8, 32);
    inv[v] = 1.0f / s;
  }
#pragma unroll
  for (int ct = 0; ct < 16; ++ct) {
#pragma unroll
    for (int v = 0; v < 8; ++v) {
      const int n = nbase + v + 8 * hi;
      __builtin_nontemporal_store(E[ct][v] * inv[v],
          Q + ((size_t)b * kHW + n) * kM + ct * 16 + lp);
    }
  }
}

extern "C" void kernel_launch(void* const* d_in, const int* in_sizes, int n_in,
                              void* d_out, int out_size, void* d_ws, size_t ws_size,
                              hipStream_t stream) {
  (void)in_sizes; (void)n_in; (void)out_size; (void)ws_size;
  const float* F = (const float*)d_in[0];
  float* out = (float*)d_out;
  float* ws = (float*)d_ws;

  float* ctr = ws;                         // [B][M][20]  = 20480 floats
  float* cn  = ws + kB * kM * kDP;         // [B][M]      = 1024 floats
  float* acc = cn + kB * kM;               // [B][M][20]  = 20480 floats
  float* Q   = out;                        // [B][N][M]
  float* cf  = out + (size_t)kB * kHW * kM;  // [B][C][M]

  k_init_centers<<<kB * kM, 32, 0, stream>>>(F, ctr, cn);
  for (int it = 0; it < 2; ++it) {
    k_zero<<<(kB * kM * kDP + 255) / 256, 256, 0, stream>>>(acc, kB * kM * kDP);
    k_assign_accum<<<ASSIGN_BLOCKS, 256, 0, stream>>>(F, ctr, cn, acc);
    k_update_centers<<<(kB * kM + 255) / 256, 256, 0, stream>>>(acc, ctr, cn);
  }
  k_assign_write<<<ASSIGN_BLOCKS, 256, 0, stream>>>(F, ctr, cn, Q);
  k_centers_feat<<<(kB * kC * kM + 255) / 256, 256, 0, stream>>>(ctr, cf);
}